// Yolov1_Loss_18219251269681
// MI455X (gfx1250) — compile-verified
//
#include <hip/hip_runtime.h>

#define NCLS_   20
#define BSZ_    16384
#define NL_     131072
#define NCELLS  (BSZ_ * 7 * 7)          // 802816 cells
#define NB1     784                     // 784*256*4 == NCELLS exactly
#define NB2     (NL_ / 256)             // 512 blocks for label pass

typedef __attribute__((ext_vector_type(2))) float v2f;
typedef __attribute__((ext_vector_type(8))) float v8f;

// Wave-level sum via V_WMMA_F32_16X16X4_F32:
//   A holds the 32 lane partials (one K slot each, other zeroed); B = all-ones 4x16,
//   so D[m][n] = partial[m] + partial[m+16] regardless of B's VGPR layout.
//   In-lane sum of the 8 D VGPRs gives half-sums; shfl_xor(16) completes the wave total.
// Then an 8-slot LDS combine produces the block total on thread 0.
__device__ __forceinline__ float block_reduce_wmma(float x, float* sm) {
    v2f a; a[0] = x;    a[1] = 0.0f;
    v2f b; b[0] = 1.0f; b[1] = 1.0f;
    v8f c = {};
    c = __builtin_amdgcn_wmma_f32_16x16x4_f32(false, a, false, b,
                                              (short)0, c, false, false);
    float s = ((c[0] + c[1]) + (c[2] + c[3])) + ((c[4] + c[5]) + (c[6] + c[7]));
    s += __shfl_xor(s, 16, 32);
    int lane = threadIdx.x & 31;
    int wid  = threadIdx.x >> 5;
    if (lane == 0) sm[wid] = s;
    __syncthreads();
    float tot = 0.0f;
    if (threadIdx.x == 0) {
        tot = ((sm[0] + sm[1]) + (sm[2] + sm[3])) +
              ((sm[4] + sm[5]) + (sm[6] + sm[7]));
    }
    return tot;                                  // valid on thread 0 only
}

// Pass 1: per-cell confidence pair. Channels 20/21 live at byte offset 80 of each
// 120-byte cell -> one aligned float2 load per cell, branchless, 4 independent
// loads per thread. Every 128B cacheline of the 96MB buffer is touched, so this
// runs at full HBM bandwidth AND warms L2 (192MB > 96MB) for the gather pass.
__global__ void noobj_kernel(const float* __restrict__ out,
                             float* __restrict__ partials) {
    __shared__ float sm[8];
    const int tid    = blockIdx.x * 256 + threadIdx.x;
    const int stride = NB1 * 256;                // 200704
    float acc = 0.0f;
#pragma unroll
    for (int k = 0; k < 4; ++k) {
        const long cell = (long)tid + (long)k * stride;     // < NCELLS by construction
        float2 cf = *(const float2*)(out + cell * 30 + 20); // 8B-aligned (120c+80)
        acc += cf.x * cf.x + cf.y * cf.y;
    }
    acc *= 0.5f;                                 // w_noobj
    float tot = block_reduce_wmma(acc, sm);
    if (threadIdx.x == 0) partials[blockIdx.x] = tot;
}

// Pass 2: one thread per label. Gathers the 120B cell (L2-resident after pass 1).
__global__ void label_kernel(const float* __restrict__ out,
                             const float* __restrict__ labels,
                             float* __restrict__ partials) {
    __shared__ float sm[8];
    int i = blockIdx.x * blockDim.x + threadIdx.x;   // NL_ % 256 == 0 -> always in range
    const float* L = labels + (long)i * 6;
    float bF = L[0], cF = L[1];
    float gx = L[2], gy = L[3], gw = L[4], gh = L[5];
    int  b = (int)bF, cc = (int)cF;
    float rowf = floorf(gx * 7.0f);
    float colf = floorf(gy * 7.0f);
    int row = (int)rowf, col = (int)colf;

    const float*  p  = out + (long)((b * 7 + row) * 7 + col) * 30;
    const float2* p2 = (const float2*)p;             // 120-byte cell base is 8B-aligned
    float pv[30];
#pragma unroll
    for (int k = 0; k < 15; ++k) { float2 t = p2[k]; pv[2 * k] = t.x; pv[2 * k + 1] = t.y; }

    // class loss vs one-hot
    float cls = 0.0f;
#pragma unroll
    for (int k = 0; k < NCLS_; ++k) {
        float d = pv[k] - (k == cc ? 1.0f : 0.0f);
        cls += d * d;
    }

    const float invS = 1.0f / 7.0f;
    float gx1 = gx - gw * 0.5f, gx2 = gx + gw * 0.5f;
    float gy1 = gy - gh * 0.5f, gy2 = gy + gh * 0.5f;
    float gArea = gw * gh;

    auto iou = [&](float bx, float by, float bw, float bh) -> float {
        float ax = (rowf + bx) * invS, ay = (colf + by) * invS;
        float ax1 = ax - bw * 0.5f, ax2 = ax + bw * 0.5f;
        float ay1 = ay - bh * 0.5f, ay2 = ay + bh * 0.5f;
        float iw = fmaxf(0.0f, fminf(ax2, gx2) - fmaxf(ax1, gx1));
        float ih = fmaxf(0.0f, fminf(ay2, gy2) - fmaxf(ay1, gy1));
        float inter = iw * ih;
        float uni = bw * bh + gArea - inter;
        return inter / (uni + 1e-6f);
    };
    float iou1 = iou(pv[22], pv[23], pv[24], pv[25]);
    float iou2 = iou(pv[26], pv[27], pv[28], pv[29]);
    bool use1 = (iou1 >= iou2);

    float gcx = gx * 7.0f - rowf, gcy = gy * 7.0f - colf;
    float xy0 = use1 ? pv[22] : pv[26], xy1 = use1 ? pv[23] : pv[27];
    float wh0 = use1 ? pv[24] : pv[28], wh1 = use1 ? pv[25] : pv[29];
    const float eps = 1e-6f;
    float d0 = xy0 - gcx, d1 = xy1 - gcy;
    float s0 = sqrtf(wh0 + eps) - sqrtf(gw + eps);
    float s1 = sqrtf(wh1 + eps) - sqrtf(gh + eps);
    float coor = 5.0f * ((d0 * d0 + d1 * d1) + (s0 * s0 + s1 * s1));

    float cp = use1 ? pv[20] : pv[21];
    float ct = use1 ? iou1 : iou2;
    float dc = cp - ct;
    float conf = dc * dc - 0.5f * cp * cp;

    float tot = block_reduce_wmma(cls + coor + conf, sm);
    if (threadIdx.x == 0) partials[blockIdx.x] = tot;
}

// Pass 3: deterministic final reduction in double, scale by 1/BSZ.
__global__ void final_kernel(const float* __restrict__ partials, int n,
                             float* __restrict__ outp) {
    __shared__ double sm[256];
    double acc = 0.0;
    for (int i = threadIdx.x; i < n; i += blockDim.x) acc += (double)partials[i];
    sm[threadIdx.x] = acc;
    __syncthreads();
    for (int s = blockDim.x >> 1; s > 0; s >>= 1) {
        if ((int)threadIdx.x < s) sm[threadIdx.x] += sm[threadIdx.x + s];
        __syncthreads();
    }
    if (threadIdx.x == 0) outp[0] = (float)(sm[0] / (double)BSZ_);
}

extern "C" void kernel_launch(void* const* d_in, const int* in_sizes, int n_in,
                              void* d_out, int out_size, void* d_ws, size_t ws_size,
                              hipStream_t stream) {
    const float* output = (const float*)d_in[0];   // (16384,7,7,30) f32
    const float* labels = (const float*)d_in[1];   // (131072,6) f32
    float* ws = (float*)d_ws;                      // NB1+NB2 = 1296 floats of scratch

    noobj_kernel<<<NB1, 256, 0, stream>>>(output, ws);
    label_kernel<<<NB2, 256, 0, stream>>>(output, labels, ws + NB1);
    final_kernel<<<1, 256, 0, stream>>>(ws, NB1 + NB2, (float*)d_out);
}